// IAMC_nonphy_30906584662135
// MI455X (gfx1250) — compile-verified
//
#include <hip/hip_runtime.h>

// f32 WMMA operand/accumulator vector types (wave32, gfx1250)
typedef __attribute__((ext_vector_type(2))) float v2f;   // A/B: 16x4 / 4x16 f32 = 2 VGPRs
typedef __attribute__((ext_vector_type(8))) float v8f;   // C/D: 16x16 f32 = 8 VGPRs

#define ROWS_PER_TILE 15    // 15 data rows + 1 anchor row per 16-row Gram tile
#define WAVES_PER_BLOCK 4   // one tile per wave; 128-thread blocks

// ---------------------------------------------------------------------------
// Kernel 1: find anchor (first label==0), pos/neg counts.
// ---------------------------------------------------------------------------
__global__ void scan_labels_kernel(const int* __restrict__ label, int B,
                                   int* __restrict__ meta) {
  __shared__ int s_min[256];
  __shared__ int s_cnt[256];
  const int tid = threadIdx.x;
  int lmin = B, lcnt = 0;
  for (int i = tid; i < B; i += 256) {
    if (label[i] == 0) { lcnt++; if (i < lmin) lmin = i; }
  }
  s_min[tid] = lmin;
  s_cnt[tid] = lcnt;
  __syncthreads();
  for (int s = 128; s > 0; s >>= 1) {
    if (tid < s) {
      s_min[tid] = min(s_min[tid], s_min[tid + s]);
      s_cnt[tid] += s_cnt[tid + s];
    }
    __syncthreads();
  }
  if (tid == 0) {
    meta[0] = s_min[0];        // anchor index (first sample of class 0)
    meta[1] = s_cnt[0] - 1;    // pos count (class 0, excluding anchor)
    meta[2] = B - s_cnt[0];    // neg count
  }
}

// ---------------------------------------------------------------------------
// Kernel 2: 4 waves per block, one 15-row tile per wave per modality.
// Gram G = A * A^T via V_WMMA_F32_16X16X4_F32 accumulated over K = D.
// Tile row 15 is the anchor row, so after the K loop:
//   G[i][i]  = ||x_i||^2,  G[i][15] = x_i . anchor,  G[15][15] = ||anchor||^2
// f32 A-operand and B-operand share register striping under transpose, so the
// same v2f register pair serves as SRC0 and SRC1.
// ---------------------------------------------------------------------------
__global__ void __launch_bounds__(32 * WAVES_PER_BLOCK)
gram_tile_kernel(const float* __restrict__ p_t,
                 const float* __restrict__ p_a,
                 const float* __restrict__ s_t,
                 const float* __restrict__ s_a,
                 const int* __restrict__ label,
                 const int* __restrict__ meta,
                 float* __restrict__ pos_part,
                 float* __restrict__ neg_part,
                 int B, int D, int tiles) {
  const int mod  = blockIdx.y;
  const float* X = (mod == 0) ? p_t : (mod == 1) ? p_a : (mod == 2) ? s_t : s_a;

  const int wave = threadIdx.x >> 5;                     // 0..3
  const int tile = blockIdx.x * WAVES_PER_BLOCK + wave;  // this wave's tile
  const bool tile_valid = (tile < tiles);
  const int anchor = meta[0];

  const int lane = threadIdx.x & 31;
  const int m    = lane & 15;     // tile row this lane feeds (A) / column (B)
  const int h    = lane >> 4;     // 0: K pair {0,1}; 1: K pair {2,3}

  // Row 15 of the tile is always the anchor; OOB / invalid tiles read row 0
  // (valid memory) and are masked out of the scoring below.
  int row;
  if (m == ROWS_PER_TILE) {
    row = anchor;
  } else {
    const int g = tile * ROWS_PER_TILE + m;
    row = (tile_valid && g < B) ? g : 0;
  }

  // Lane streams exactly the K-pair it owns: elements (4t+2h, 4t+2h+1).
  const v2f* __restrict__ rp = (const v2f*)(X + (size_t)row * D);
  v8f c = {};
  const int nchunk = D >> 2;      // K chunks of 4
#pragma unroll 16
  for (int t = 0; t < nchunk; ++t) {
    v2f a = rp[2 * t + h];        // clause of 16 coalesced b64 loads
    // D = A(16x4) * A^T(4x16) + C  -> Gram accumulation
    c = __builtin_amdgcn_wmma_f32_16x16x4_f32(
        /*neg_a=*/false, a, /*neg_b=*/false, a,
        /*c_mod=*/(short)0, c, /*reuse_a=*/false, /*reuse_b=*/false);
  }

  // Spill the 16x16 Gram tile: VGPR r holds M=r (lanes 0-15) / M=r+8 (16-31).
  __shared__ float gm[WAVES_PER_BLOCK][16][16];
#pragma unroll
  for (int r = 0; r < 8; ++r) {
    gm[wave][r + 8 * h][m] = c[r];
  }
  __syncthreads();

  __shared__ float posv[WAVES_PER_BLOCK][16];
  __shared__ float negv[WAVES_PER_BLOCK][16];
  if (lane < 16) {
    float pv = 0.0f, nv = 0.0f;
    const int j = lane;
    if (tile_valid && j < ROWS_PER_TILE) {
      const int g = tile * ROWS_PER_TILE + j;
      if (g < B) {
        const float sq  = gm[wave][j][j];
        const float dt  = gm[wave][j][ROWS_PER_TILE];
        const float asq = gm[wave][ROWS_PER_TILE][ROWS_PER_TILE];
        const float cosv = dt / sqrtf(sq * asq);
        const int lb = label[g];
        if (lb == 0) {
          if (g != anchor) pv = cosv;   // pos_mask
        } else {
          nv = cosv;                    // neg_mask
        }
      }
    }
    posv[wave][j] = pv;
    negv[wave][j] = nv;
  }
  __syncthreads();

  if (lane == 0 && tile_valid) {
    float ps = 0.0f, ns = 0.0f;
    for (int j = 0; j < ROWS_PER_TILE; ++j) {
      ps += posv[wave][j];
      ns += negv[wave][j];
    }
    pos_part[mod * tiles + tile] = ps;   // fixed slot -> deterministic
    neg_part[mod * tiles + tile] = ns;
  }
}

// ---------------------------------------------------------------------------
// Kernel 3: deterministic fixed-order reduction of per-tile partials + final
// loss. 8 groups (4 modalities x pos/neg), 32 lanes each stride the partials,
// then thread 0 folds everything in a fixed order.
// ---------------------------------------------------------------------------
__global__ void finalize_kernel(const float* __restrict__ pos_part,
                                const float* __restrict__ neg_part,
                                const int* __restrict__ meta,
                                float* __restrict__ out, int tiles) {
  __shared__ float s[256];
  const int tid = threadIdx.x;
  const int grp = tid >> 5;   // 0..7: (mod<<1) | is_neg
  const int ln  = tid & 31;
  const int mod = grp >> 1;
  const float* src = (grp & 1) ? (neg_part + mod * tiles)
                               : (pos_part + mod * tiles);
  float acc = 0.0f;
  for (int t = ln; t < tiles; t += 32) acc += src[t];
  s[tid] = acc;
  __syncthreads();

  if (tid == 0) {
    float sums[8];
    for (int g = 0; g < 8; ++g) {
      float a = 0.0f;
      for (int l = 0; l < 32; ++l) a += s[g * 32 + l];
      sums[g] = a;
    }
    const int pc = meta[1];
    const int nc = meta[2];
    float total = 0.0f;
    for (int mdl = 0; mdl < 4; ++mdl) {
      const float pos_score = (pc > 0) ? sums[2 * mdl]     / (float)max(pc, 1) : 0.0f;
      const float neg_score = (nc > 0) ? sums[2 * mdl + 1] / (float)max(nc, 1) : 0.0f;
      total += 2.0f - pos_score + neg_score;
    }
    out[0] = total * 0.25f;
  }
}

// ---------------------------------------------------------------------------
extern "C" void kernel_launch(void* const* d_in, const int* in_sizes, int n_in,
                              void* d_out, int out_size, void* d_ws, size_t ws_size,
                              hipStream_t stream) {
  const int*   label = (const int*)d_in[0];
  const float* p_t   = (const float*)d_in[1];
  const float* p_a   = (const float*)d_in[2];
  const float* s_t   = (const float*)d_in[3];
  const float* s_a   = (const float*)d_in[4];

  const int B = in_sizes[0];
  const int D = in_sizes[1] / B;
  const int tiles = (B + ROWS_PER_TILE - 1) / ROWS_PER_TILE;

  // Workspace layout: [meta: 3 ints][pad to 64B][pos_part 4*tiles][neg_part 4*tiles]
  char* ws = (char*)d_ws;
  int*   meta     = (int*)ws;
  float* pos_part = (float*)(ws + 64);
  float* neg_part = pos_part + 4 * tiles;

  scan_labels_kernel<<<1, 256, 0, stream>>>(label, B, meta);

  dim3 grid((tiles + WAVES_PER_BLOCK - 1) / WAVES_PER_BLOCK, 4, 1);
  gram_tile_kernel<<<grid, 32 * WAVES_PER_BLOCK, 0, stream>>>(
      p_t, p_a, s_t, s_a, label, meta, pos_part, neg_part, B, D, tiles);

  finalize_kernel<<<1, 256, 0, stream>>>(pos_part, neg_part, meta,
                                         (float*)d_out, tiles);
}